// LinearAttention_25795573579902
// MI455X (gfx1250) — compile-verified
//
#include <hip/hip_runtime.h>
#include <hip/hip_bf16.h>
#include <math.h>

typedef float v2f __attribute__((ext_vector_type(2)));
typedef float v8f __attribute__((ext_vector_type(8)));

#define HEADS 4
#define DHEAD 32
#define CCH   128      // input/output channels C
#define HID   128      // HEADS*DHEAD
#define NN    65536    // T*F
#define BATCH 8
#define TN    64       // n-positions per subtile
#define NSUB  8        // subtiles per K1 workgroup (amortizes S/Z atomics 8x)
#define PITCH 72       // LDS row pitch in floats (bank-stagger)
#define LN_EPS 1e-5f

__device__ __forceinline__ v8f wmma_f32_16x16x4(v2f a, v2f b, v8f c) {
  // D(16x16,f32) = A(16x4,f32) x B(4x16,f32) + C
  return __builtin_amdgcn_wmma_f32_16x16x4_f32(false, a, false, b, (short)0, c,
                                               false, false);
}

// ---------------------------------------------------------------------------
// K1: k/v rows of the qkv projection (WMMA) + exp(k) + context partials S,Z.
// Each workgroup owns NSUB*TN = 512 n-positions; context/Z partials live in
// registers across subtiles and are flushed with one atomic pass at the end.
// q rows are NOT computed here -- K3 recomputes them from x, eliminating the
// 537 MB q round-trip through HBM at zero extra FLOPs.
// ---------------------------------------------------------------------------
__global__ __launch_bounds__(256)
void k1_kv_ctx(const float* __restrict__ x,
               const float* __restrict__ w_qkv,
               const float* __restrict__ b_qkv,
               float* __restrict__ S_ws,
               float* __restrict__ Z_ws) {
  extern __shared__ float smem[];
  float* xs = smem;                  // [128][PITCH] x tile
  float* kv = smem + CCH * PITCH;    // [256][PITCH] k (rows 0-127), v (128-255)

  const int tid    = threadIdx.x;
  const int lane   = tid & 31;
  const int wave   = tid >> 5;
  const int l15    = lane & 15;
  const int khalf  = (lane >> 4) * 2;   // A/B: k = 4*ks + khalf + {0,1}
  const int rowAdd = 8 * (lane >> 4);   // C/D: M = g + 8*(lane/16)
  const int b = blockIdx.y;

  const float* xb = x + (size_t)b * CCH * NN;

  // per-wave context assignment: 8 waves = 4 heads x 2 d-tiles
  const int ch  = wave >> 1;            // head
  const int cmt = wave & 1;             // d-tile (rows cmt*16..+15)
  v8f ctx0 = {0.f,0.f,0.f,0.f,0.f,0.f,0.f,0.f};   // e = 0..15
  v8f ctx1 = {0.f,0.f,0.f,0.f,0.f,0.f,0.f,0.f};   // e = 16..31
  float zsum = 0.f;

  for (int it = 0; it < NSUB; ++it) {
    const int n0 = (blockIdx.x * NSUB + it) * TN;
    __syncthreads();   // prior subtile's kv readers done before we overwrite

    // stage x tile (b128, coalesced along n)
    for (int i = tid; i < CCH * TN / 4; i += 256) {
      const int c = i >> 4, j4 = (i & 15) * 4;
      *(float4*)(xs + c * PITCH + j4) =
          *(const float4*)(xb + (size_t)c * NN + n0 + j4);
    }
    __syncthreads();

    // kv = W_qkv[128:384] @ x(128xTN); each wave owns 32 rows (2 M-tiles).
    // A operand hoisted into registers, reused across the 4 N-tiles.
    for (int mt = 0; mt < 2; ++mt) {
      const int mBase = wave * 32 + mt * 16;              // kv-row base
      const float* Wr = w_qkv + (size_t)(HID + mBase + l15) * CCH;
      v2f a[32];
      #pragma unroll
      for (int ks = 0; ks < 32; ++ks) {
        a[ks].x = Wr[ks * 4 + khalf];
        a[ks].y = Wr[ks * 4 + khalf + 1];
      }
      for (int nt = 0; nt < 4; ++nt) {
        const int nB = nt * 16;
        v8f acc = {0.f,0.f,0.f,0.f,0.f,0.f,0.f,0.f};
        #pragma unroll
        for (int ks = 0; ks < 32; ++ks) {
          const int k0 = ks * 4 + khalf;
          v2f bb;
          bb.x = xs[k0 * PITCH + nB + l15];
          bb.y = xs[(k0 + 1) * PITCH + nB + l15];
          acc = wmma_f32_16x16x4(a[ks], bb, acc);
        }
        const int col = nB + l15;
        #pragma unroll
        for (int g = 0; g < 8; ++g) {
          const int m = mBase + g + rowAdd;
          kv[m * PITCH + col] = acc[g] + b_qkv[HID + m];
        }
      }
    }
    __syncthreads();

    // exp(k) in place (no max-sub: k ~ N(0,1), fp32-safe, same math as the
    // reference softmax since we divide by Z = sum(exp) in K2)
    for (int i = tid; i < HID * TN; i += 256) {
      const int r = i >> 6, j = i & 63;
      float* p = kv + r * PITCH + j;
      *p = __expf(*p);
    }
    __syncthreads();

    // Z[h*32+d] partial (register accumulate across subtiles)
    if (tid < HID) {
      const float* kr = kv + tid * PITCH;
      float s = 0.f;
      for (int j = 0; j < TN; ++j) s += kr[j];
      zsum += s;
    }

    // context partial: S[d][e] += exp(k)(32xTN) @ v^T(TNx32), per head
    {
      const float* ka = kv + (ch * DHEAD + cmt * 16 + l15) * PITCH;
      const float* v0 = kv + (HID + ch * DHEAD + l15) * PITCH;
      const float* v1 = v0 + 16 * PITCH;
      #pragma unroll
      for (int ks = 0; ks < TN / 4; ++ks) {
        const int k0 = ks * 4 + khalf;
        v2f a, b0, b1;
        a.x  = ka[k0];     a.y  = ka[k0 + 1];
        b0.x = v0[k0];     b0.y = v0[k0 + 1];   // B[k=n][e] = v[e][n]
        b1.x = v1[k0];     b1.y = v1[k0 + 1];
        ctx0 = wmma_f32_16x16x4(a, b0, ctx0);
        ctx1 = wmma_f32_16x16x4(a, b1, ctx1);
      }
    }
  }

  // flush partials (one atomic pass per workgroup)
  if (tid < HID) atomicAdd(&Z_ws[b * HID + tid], zsum);
  {
    float* Sb = S_ws + ((size_t)b * HEADS + ch) * DHEAD * DHEAD;
    #pragma unroll
    for (int g = 0; g < 8; ++g) {
      const int d = cmt * 16 + g + rowAdd;
      atomicAdd(&Sb[d * DHEAD + l15],      ctx0[g]);
      atomicAdd(&Sb[d * DHEAD + 16 + l15], ctx1[g]);
    }
  }
}

// ---------------------------------------------------------------------------
// K2: ctxT[b,h,e,d] = S[b,h,d,e] / (Z[b,h,d] * N)   (folds v/N scaling)
// ---------------------------------------------------------------------------
__global__ void k2_ctx_finalize(const float* __restrict__ S_ws,
                                const float* __restrict__ Z_ws,
                                float* __restrict__ ctxT) {
  const int idx = blockIdx.x * 256 + threadIdx.x;
  if (idx >= BATCH * HEADS * DHEAD * DHEAD) return;
  const int e  = idx & 31;
  const int d  = (idx >> 5) & 31;
  const int bh = idx >> 10;                          // b*HEADS + h
  const float z = Z_ws[(bh >> 2) * HID + (bh & 3) * DHEAD + d];
  ctxT[(bh * DHEAD + e) * DHEAD + d] = S_ws[idx] / (z * (float)NN);
}

// ---------------------------------------------------------------------------
// K3: q rows of qkv projection from x, softmax(d), attn = ctxT @ q,
//     out = W_out @ attn + b_out, channel LayerNorm.
// LDS ping-pong: xs holds x then attn; qs holds q then out.
// ---------------------------------------------------------------------------
__global__ __launch_bounds__(256)
void k3_out(const float* __restrict__ x,
            const float* __restrict__ w_qkv,
            const float* __restrict__ b_qkv,
            const float* __restrict__ ctxT,
            const float* __restrict__ w_out,
            const float* __restrict__ b_out,
            const float* __restrict__ g_ln,
            float* __restrict__ out) {
  extern __shared__ float smem[];
  float* xs  = smem;                     // [128][PITCH] x tile -> attn tile
  float* qs  = smem + CCH * PITCH;       // [128][PITCH] q tile -> out tile
  float* red = smem + 2 * CCH * PITCH;   // 640 floats: partials + mean/invstd

  const int tid    = threadIdx.x;
  const int lane   = tid & 31;
  const int wave   = tid >> 5;
  const int l15    = lane & 15;
  const int khalf  = (lane >> 4) * 2;
  const int rowAdd = 8 * (lane >> 4);
  const int b  = blockIdx.y;
  const int n0 = blockIdx.x * TN;

  // stage x tile
  const float* xb = x + (size_t)b * CCH * NN + n0;
  for (int i = tid; i < CCH * TN / 4; i += 256) {
    const int r = i >> 4, j4 = (i & 15) * 4;
    *(float4*)(xs + r * PITCH + j4) =
        *(const float4*)(xb + (size_t)r * NN + j4);
  }
  __syncthreads();

  // q = W_qkv[0:128] @ x(128xTN) + b; each wave owns 16 rows (1 M-tile)
  {
    const int mBase = wave * 16;
    const float* Wr = w_qkv + (size_t)(mBase + l15) * CCH;
    v2f a[32];
    #pragma unroll
    for (int ks = 0; ks < 32; ++ks) {
      a[ks].x = Wr[ks * 4 + khalf];
      a[ks].y = Wr[ks * 4 + khalf + 1];
    }
    for (int nt = 0; nt < 4; ++nt) {
      v8f acc = {0.f,0.f,0.f,0.f,0.f,0.f,0.f,0.f};
      #pragma unroll
      for (int ks = 0; ks < 32; ++ks) {
        const int k0 = ks * 4 + khalf;
        v2f bb;
        bb.x = xs[k0 * PITCH + nt * 16 + l15];
        bb.y = xs[(k0 + 1) * PITCH + nt * 16 + l15];
        acc = wmma_f32_16x16x4(a[ks], bb, acc);
      }
      const int col = nt * 16 + l15;
      #pragma unroll
      for (int g = 0; g < 8; ++g) {
        const int m = mBase + g + rowAdd;
        qs[m * PITCH + col] = acc[g] + b_qkv[m];
      }
    }
  }
  __syncthreads();

  // q softmax over d, * DHEAD^-0.5 (column cached in registers: one exp)
  {
    const int hh = tid >> 6, j = tid & 63;   // 4 heads x 64 cols
    float* col = qs + (hh * DHEAD) * PITCH + j;
    float v[DHEAD];
    float mx = -1e30f;
    #pragma unroll
    for (int d = 0; d < DHEAD; ++d) {
      v[d] = col[d * PITCH];
      mx = fmaxf(mx, v[d]);
    }
    float s = 0.f;
    #pragma unroll
    for (int d = 0; d < DHEAD; ++d) { v[d] = __expf(v[d] - mx); s += v[d]; }
    const float inv = 0.17677669529663687f / s;  // 32^-0.5 / sum
    #pragma unroll
    for (int d = 0; d < DHEAD; ++d) col[d * PITCH] = v[d] * inv;
  }
  __syncthreads();

  // attn_h(32xTN) = ctxT_h(32x32) @ q_h(32xTN) -> overwrite xs
  {
    const int h     = wave >> 1;
    const int mt    = wave & 1;
    const int mBase = mt * 16;
    const float* A = ctxT + ((size_t)b * HEADS + h) * DHEAD * DHEAD; // [e][d]
    v2f ac[8];
    #pragma unroll
    for (int ks = 0; ks < 8; ++ks) {
      ac[ks].x = A[(mBase + l15) * DHEAD + ks * 4 + khalf];
      ac[ks].y = A[(mBase + l15) * DHEAD + ks * 4 + khalf + 1];
    }
    for (int nt = 0; nt < 4; ++nt) {
      v8f acc = {0.f,0.f,0.f,0.f,0.f,0.f,0.f,0.f};
      #pragma unroll
      for (int ks = 0; ks < 8; ++ks) {
        const int k0 = ks * 4 + khalf;
        v2f bb;
        bb.x = qs[(h * DHEAD + k0) * PITCH + nt * 16 + l15];
        bb.y = qs[(h * DHEAD + k0 + 1) * PITCH + nt * 16 + l15];
        acc = wmma_f32_16x16x4(ac[ks], bb, acc);
      }
      const int col = nt * 16 + l15;
      #pragma unroll
      for (int g = 0; g < 8; ++g) {
        const int e = mBase + g + rowAdd;
        xs[(h * DHEAD + e) * PITCH + col] = acc[g];
      }
    }
  }
  __syncthreads();

  // out = W_out(128x128) @ attn(128xTN) + b_out -> overwrite qs
  {
    const int mBase = wave * 16;
    const float* Wr = w_out + (size_t)(mBase + l15) * HID;
    v2f a[32];
    #pragma unroll
    for (int ks = 0; ks < 32; ++ks) {
      a[ks].x = Wr[ks * 4 + khalf];
      a[ks].y = Wr[ks * 4 + khalf + 1];
    }
    for (int nt = 0; nt < 4; ++nt) {
      v8f acc = {0.f,0.f,0.f,0.f,0.f,0.f,0.f,0.f};
      #pragma unroll
      for (int ks = 0; ks < 32; ++ks) {
        const int k0 = ks * 4 + khalf;
        v2f bb;
        bb.x = xs[k0 * PITCH + nt * 16 + l15];
        bb.y = xs[(k0 + 1) * PITCH + nt * 16 + l15];
        acc = wmma_f32_16x16x4(a[ks], bb, acc);
      }
      const int col = nt * 16 + l15;
      #pragma unroll
      for (int g = 0; g < 8; ++g) {
        const int o = mBase + g + rowAdd;
        qs[o * PITCH + col] = acc[g] + b_out[o];
      }
    }
  }
  __syncthreads();

  // LayerNorm over 128 channels per column (biased variance)
  float* psum = red;          // [4][64]
  float* psq  = red + 256;    // [4][64]
  float* mInv = red + 512;    // [2][64]
  {
    const int j = tid & 63, seg = tid >> 6;
    float s = 0.f, sq = 0.f;
    for (int r = seg * 32; r < seg * 32 + 32; ++r) {
      const float v = qs[r * PITCH + j];
      s += v; sq += v * v;
    }
    psum[seg * 64 + j] = s;
    psq [seg * 64 + j] = sq;
  }
  __syncthreads();
  if (tid < 64) {
    const float s  = psum[tid] + psum[64+tid] + psum[128+tid] + psum[192+tid];
    const float sq = psq [tid] + psq [64+tid] + psq [128+tid] + psq [192+tid];
    const float mean = s * (1.0f / 128.0f);
    const float var  = sq * (1.0f / 128.0f) - mean * mean;
    mInv[tid]      = mean;
    mInv[64 + tid] = rsqrtf(var + LN_EPS);
  }
  __syncthreads();
  float* ob = out + (size_t)b * CCH * NN + n0;
  for (int i = tid; i < CCH * TN / 4; i += 256) {
    const int r = i >> 4, j4 = (i & 15) * 4;
    const float g = g_ln[r];
    float4 vv = *(const float4*)(qs + r * PITCH + j4);
    float4 ov;
    ov.x = (vv.x - mInv[j4 + 0]) * mInv[64 + j4 + 0] * g;
    ov.y = (vv.y - mInv[j4 + 1]) * mInv[64 + j4 + 1] * g;
    ov.z = (vv.z - mInv[j4 + 2]) * mInv[64 + j4 + 2] * g;
    ov.w = (vv.w - mInv[j4 + 3]) * mInv[64 + j4 + 3] * g;
    *(float4*)(ob + (size_t)r * NN + j4) = ov;
  }
}

// ---------------------------------------------------------------------------
extern "C" void kernel_launch(void* const* d_in, const int* in_sizes, int n_in,
                              void* d_out, int out_size, void* d_ws,
                              size_t ws_size, hipStream_t stream) {
  const float* x     = (const float*)d_in[0];
  const float* w_qkv = (const float*)d_in[1];
  const float* b_qkv = (const float*)d_in[2];
  const float* w_out = (const float*)d_in[3];
  const float* b_out = (const float*)d_in[4];
  const float* g_ln  = (const float*)d_in[5];
  float* outp = (float*)d_out;

  // workspace layout: S (128 KB) | Z (4 KB) | ctxT (128 KB)  (~260 KB total)
  float* S_ws = (float*)d_ws;
  float* Z_ws = S_ws + BATCH * HEADS * DHEAD * DHEAD;
  float* ctxT = Z_ws + BATCH * HID;

  hipMemsetAsync(S_ws, 0,
                 (BATCH * HEADS * DHEAD * DHEAD + BATCH * HID) * sizeof(float),
                 stream);

  dim3 blk(256);
  dim3 grid1(NN / (TN * NSUB), BATCH);   // 128 x 8
  const size_t sm1 = (size_t)(CCH + 2 * HID) * PITCH * sizeof(float); // 110592
  k1_kv_ctx<<<grid1, blk, sm1, stream>>>(x, w_qkv, b_qkv, S_ws, Z_ws);

  const int ctxN = BATCH * HEADS * DHEAD * DHEAD;
  k2_ctx_finalize<<<(ctxN + 255) / 256, 256, 0, stream>>>(S_ws, Z_ws, ctxT);

  dim3 grid3(NN / TN, BATCH);            // 1024 x 8
  const size_t sm3 = (size_t)(2 * CCH * PITCH + 640) * sizeof(float);
  k3_out<<<grid3, blk, sm3, stream>>>(x, w_qkv, b_qkv, ctxT, w_out, b_out,
                                      g_ln, outp);
}